// GNNModel_62319975465020
// MI455X (gfx1250) — compile-verified
//
#include <hip/hip_runtime.h>
#include <math.h>

#ifndef __has_builtin
#define __has_builtin(x) 0
#endif
#define USE_WMMA_F32 __has_builtin(__builtin_amdgcn_wmma_f32_16x16x4_f32)

typedef __attribute__((ext_vector_type(2))) float v2f;
typedef __attribute__((ext_vector_type(8))) float v8f;

#define HEADS 5
#define NBLK 4   // N-tiles (16 cols each) per wave -> 16x64 output strip

__device__ __forceinline__ float softplusf(float x) {
    return fmaxf(x, 0.f) + log1pf(expf(-fabsf(x)));
}
__device__ __forceinline__ float lrelu_f(float x, float s) { return x >= 0.f ? x : s * x; }

// float atomic max via ordered-int trick (works for mixed signs, init -inf)
__device__ __forceinline__ void atomicMaxF(float* addr, float v) {
    if (v >= 0.f) atomicMax((int*)addr, __float_as_int(v));
    else          atomicMin((unsigned int*)addr, __float_as_uint(v));
}

// ---------------------------------------------------------------------------
// WMMA f32 GEMM: C[M,Nout] = A[M,K] @ B[K,Nout] (+bias, epilogue)
// One wave per 16x(16*NBLK) output strip; K stepped by 4 via
// V_WMMA_F32_16X16X4_F32. A loaded once per k-step, 4 WMMAs issued on it.
// Out-of-range columns are clamped (not zeroed): D column n depends only on
// B data in lanes with lane%16==n, so clamped lanes only produce garbage in
// columns that are never stored.
// EPI: 0 = bias only, 1 = bias+softplus, 2 = bias+sigmoid
// ---------------------------------------------------------------------------
template <int EPI>
__global__ __launch_bounds__(256) void wmma_gemm(
    const float* __restrict__ A, int lda,
    const float* __restrict__ B, int ldb,
    const float* __restrict__ bias,
    float* __restrict__ C, int ldc,
    int M, int K, int Nout, int tiles_nb, int total_waves)
{
    int wave = blockIdx.x * (blockDim.x >> 5) + (threadIdx.x >> 5);
    if (wave >= total_waves) return;              // wave-uniform exit: EXEC stays all-1s
    int mt  = wave / tiles_nb;
    int ntb = wave - mt * tiles_nb;                // block of NBLK n-tiles
    int lane = threadIdx.x & 31;
    int hi = lane >> 4;                            // half-wave: selects K pair
    int lr = lane & 15;                            // A row / B,D column
    int arow = mt * 16 + lr; if (arow >= M) arow = M - 1;

    int  col[NBLK];
    bool cok[NBLK];
    const float* Bp[NBLK];
#pragma unroll
    for (int j = 0; j < NBLK; ++j) {
        col[j] = (ntb * NBLK + j) * 16 + lr;
        cok[j] = col[j] < Nout;
        Bp[j]  = B + (size_t)(2 * hi) * ldb + (cok[j] ? col[j] : 0);
    }
    const float* Ap = A + (size_t)arow * lda + 2 * hi;     // A[arow, k + 2*hi + {0,1}]

#if USE_WMMA_F32
    v8f acc[NBLK];
#pragma unroll
    for (int j = 0; j < NBLK; ++j) acc[j] = (v8f){};
    for (int k = 0; k < K; k += 4) {
        float2 av = *(const float2*)(Ap + k);              // 8B-aligned (lda even, k%4==0)
        v2f a; a.x = av.x; a.y = av.y;
#pragma unroll
        for (int j = 0; j < NBLK; ++j) {
            v2f b;
            b.x = Bp[j][(size_t)k * ldb];
            b.y = Bp[j][(size_t)(k + 1) * ldb];
            acc[j] = __builtin_amdgcn_wmma_f32_16x16x4_f32(
                false, a, false, b, (short)0, acc[j], false, false);
        }
    }
#pragma unroll
    for (int j = 0; j < NBLK; ++j) {
        float bb = (bias && cok[j]) ? bias[col[j]] : 0.f;
#pragma unroll
        for (int v = 0; v < 8; ++v) {
            int row = mt * 16 + hi * 8 + v;               // D layout: M = v + 8*hi
            if (cok[j] && row < M) {
                float y = acc[j][v] + bb;
                if (EPI == 1) y = softplusf(y);
                else if (EPI == 2) y = 1.f / (1.f + expf(-y));
                C[(size_t)row * ldc + col[j]] = y;
            }
        }
    }
#else
    // scalar fallback (kept only as compile insurance; WMMA path is live)
    for (int j = 0; j < NBLK; ++j) {
        float accs[8];
#pragma unroll
        for (int v = 0; v < 8; ++v) accs[v] = 0.f;
        for (int k = 0; k < K; ++k) {
            float bv = cok[j] ? B[(size_t)k * ldb + col[j]] : 0.f;
#pragma unroll
            for (int v = 0; v < 8; ++v) {
                int row = mt * 16 + hi * 8 + v; if (row >= M) row = M - 1;
                accs[v] += A[(size_t)row * lda + k] * bv;
            }
        }
        float bb = (bias && cok[j]) ? bias[col[j]] : 0.f;
#pragma unroll
        for (int v = 0; v < 8; ++v) {
            int row = mt * 16 + hi * 8 + v;
            if (cok[j] && row < M) {
                float y = accs[v] + bb;
                if (EPI == 1) y = softplusf(y);
                else if (EPI == 2) y = 1.f / (1.f + expf(-y));
                C[(size_t)row * ldc + col[j]] = y;
            }
        }
    }
#endif
}

// ---------------------------------------------------------------------------
// Elementwise / reduction / scatter kernels
// ---------------------------------------------------------------------------
__global__ void fill_f32(float* __restrict__ p, float v, size_t n) {
    size_t t = (size_t)blockIdx.x * blockDim.x + threadIdx.x;
    if (t < n) p[t] = v;
}

__global__ void init_bias2d(float* __restrict__ out, const float* __restrict__ b,
                            int cmask, size_t total) {
    size_t t = (size_t)blockIdx.x * blockDim.x + threadIdx.x;
    if (t < total) out[t] = b[t & cmask];
}

__global__ void deg_accum(const int* __restrict__ ei, int E, float* __restrict__ deg) {
    int t = blockIdx.x * blockDim.x + threadIdx.x;
    if (t < E) atomicAdd(&deg[ei[E + t]], 1.0f);
}

__global__ void deg_to_dinv(float* __restrict__ d, int Nn) {
    int t = blockIdx.x * blockDim.x + threadIdx.x;
    if (t < Nn) { float v = d[t]; d[t] = v > 0.f ? rsqrtf(v) : 0.f; }
}

// out[dst,:] += xw[src,:] * dinv[src]*dinv[dst]   (C = 1<<cshift)
__global__ void gcn_scatter(const int* __restrict__ ei, int E, int Ep,
                            const float* __restrict__ dinv, const float* __restrict__ xw,
                            float* __restrict__ out, int cshift) {
    long long t = (long long)blockIdx.x * blockDim.x + threadIdx.x;
    if (t >= ((long long)Ep << cshift)) return;
    int e = (int)(t >> cshift);
    int c = (int)(t & ((1 << cshift) - 1));
    int src = e < E ? ei[e]     : e - E;
    int dst = e < E ? ei[E + e] : e - E;
    float norm = dinv[src] * dinv[dst];
    atomicAdd(out + (((size_t)dst) << cshift) + c,
              xw[(((size_t)src) << cshift) + c] * norm);
}

// a_s[n,h] = sum_c xl[n,h*C+c]*a_src[h,c]; a_d likewise
__global__ void gat_attn(const float* __restrict__ xl, const float* __restrict__ a_src,
                         const float* __restrict__ a_dst, float* __restrict__ as_,
                         float* __restrict__ ad_, int Nn, int C) {
    int t = blockIdx.x * blockDim.x + threadIdx.x;
    if (t >= Nn * HEADS) return;
    int n = t / HEADS, h = t - n * HEADS;
    const float* xr = xl + (size_t)n * HEADS * C + (size_t)h * C;
    const float* s0 = a_src + h * C;
    const float* s1 = a_dst + h * C;
    float v0 = 0.f, v1 = 0.f;
    for (int c = 0; c < C; ++c) { float x = xr[c]; v0 += x * s0[c]; v1 += x * s1[c]; }
    as_[t] = v0; ad_[t] = v1;
}

__global__ void gat_e1(const int* __restrict__ ei, int E, int Ep,
                       const float* __restrict__ as_, const float* __restrict__ ad_,
                       float* __restrict__ eb, float* __restrict__ m) {
    int t = blockIdx.x * blockDim.x + threadIdx.x;
    if (t >= Ep * HEADS) return;
    int e = t / HEADS, h = t - e * HEADS;
    int src = e < E ? ei[e]     : e - E;
    int dst = e < E ? ei[E + e] : e - E;
    float v = lrelu_f(as_[src * HEADS + h] + ad_[dst * HEADS + h], 0.2f);
    eb[t] = v;
    atomicMaxF(&m[dst * HEADS + h], v);
}

__global__ void gat_e2(const int* __restrict__ ei, int E, int Ep,
                       float* __restrict__ eb, const float* __restrict__ m,
                       float* __restrict__ s) {
    int t = blockIdx.x * blockDim.x + threadIdx.x;
    if (t >= Ep * HEADS) return;
    int e = t / HEADS, h = t - e * HEADS;
    int dst = e < E ? ei[E + e] : e - E;
    float pe = expf(eb[t] - m[dst * HEADS + h]);
    eb[t] = pe;
    atomicAdd(&s[dst * HEADS + h], pe);
}

// acc[dst,c] += sum_h xl[src,h*C+c] * (pe[e,h]/s[dst,h])   (head-sum folded in)
__global__ void gat_e3(const int* __restrict__ ei, int E, int Ep,
                       const float* __restrict__ eb, const float* __restrict__ s,
                       const float* __restrict__ xl, float* __restrict__ acc, int C) {
    long long t = (long long)blockIdx.x * blockDim.x + threadIdx.x;
    if (t >= (long long)Ep * C) return;
    int e = (int)(t / C);
    int c = (int)(t - (long long)e * C);
    int src = e < E ? ei[e]     : e - E;
    int dst = e < E ? ei[E + e] : e - E;
    float sum = 0.f;
#pragma unroll
    for (int h = 0; h < HEADS; ++h) {
        float alpha = eb[e * HEADS + h] / s[dst * HEADS + h];
        sum += xl[(size_t)src * HEADS * C + (size_t)h * C + c] * alpha;
    }
    atomicAdd(&acc[(size_t)dst * C + c], sum);
}

// out[n,c] = (acc[n,c]/HEADS + b[c]) [optionally lrelu 0.01]
__global__ void gat_fin(const float* __restrict__ acc, const float* __restrict__ b,
                        float* __restrict__ out, int ldo, int C, int Nn, int do_lrelu) {
    long long t = (long long)blockIdx.x * blockDim.x + threadIdx.x;
    if (t >= (long long)Nn * C) return;
    int n = (int)(t / C);
    int c = (int)(t - (long long)n * C);
    float v = acc[t] * (1.f / HEADS) + b[c];
    if (do_lrelu) v = lrelu_f(v, 0.01f);
    out[(size_t)n * ldo + c] = v;
}

// BatchNorm1d (training stats, biased var) + leaky_relu(0.01); one block per column
__global__ void bn_lrelu(const float* __restrict__ in, int ldi,
                         const float* __restrict__ g, const float* __restrict__ b,
                         float* __restrict__ outA, int ldA,
                         float* __restrict__ outB, int ldB, int Nrows) {
    __shared__ float r0[256], r1[256];
    int col = blockIdx.x;
    float s = 0.f, sq = 0.f;
    for (int r = threadIdx.x; r < Nrows; r += blockDim.x) {
        float v = in[(size_t)r * ldi + col];
        s += v; sq += v * v;
    }
    r0[threadIdx.x] = s; r1[threadIdx.x] = sq;
    __syncthreads();
    for (int off = 128; off > 0; off >>= 1) {
        if (threadIdx.x < off) { r0[threadIdx.x] += r0[threadIdx.x + off];
                                 r1[threadIdx.x] += r1[threadIdx.x + off]; }
        __syncthreads();
    }
    float mean = r0[0] / Nrows;
    float var  = r1[0] / Nrows - mean * mean;
    float rstd = rsqrtf(var + 1e-5f);
    float gc = g[col], bc = b[col];
    for (int r = threadIdx.x; r < Nrows; r += blockDim.x) {
        float v = (in[(size_t)r * ldi + col] - mean) * rstd * gc + bc;
        v = lrelu_f(v, 0.01f);
        if (outA) outA[(size_t)r * ldA + col] = v;
        if (outB) outB[(size_t)r * ldB + col] = v;
    }
}

// ---------------------------------------------------------------------------
// Host
// ---------------------------------------------------------------------------
static inline void gemm(int epi, const float* A, int lda, const float* B, int ldb,
                        const float* bias, float* C, int ldc,
                        int M, int K, int Nout, hipStream_t st) {
    int tiles_m  = (M + 15) >> 4;
    int tiles_n  = (Nout + 15) >> 4;
    int tiles_nb = (tiles_n + NBLK - 1) / NBLK;
    int total    = tiles_m * tiles_nb;
    int blocks   = (total + 7) / 8;            // 8 waves per 256-thread block
    if (epi == 0)      wmma_gemm<0><<<blocks, 256, 0, st>>>(A, lda, B, ldb, bias, C, ldc, M, K, Nout, tiles_nb, total);
    else if (epi == 1) wmma_gemm<1><<<blocks, 256, 0, st>>>(A, lda, B, ldb, bias, C, ldc, M, K, Nout, tiles_nb, total);
    else               wmma_gemm<2><<<blocks, 256, 0, st>>>(A, lda, B, ldb, bias, C, ldc, M, K, Nout, tiles_nb, total);
}

#define B256(n) ((unsigned)(((n) + 255) / 256))

extern "C" void kernel_launch(void* const* d_in, const int* in_sizes, int n_in,
                              void* d_out, int out_size, void* d_ws, size_t ws_size,
                              hipStream_t stream) {
    const float* x  = (const float*)d_in[0];
    const int*   ei = (const int*)d_in[1];
    // params in setup_inputs() insertion order
    const float* w1  = (const float*)d_in[2];  const float* b1  = (const float*)d_in[3];
    const float* w2  = (const float*)d_in[4];  const float* b2  = (const float*)d_in[5];
    const float* w3  = (const float*)d_in[6];  const float* b3  = (const float*)d_in[7];
    const float* g0  = (const float*)d_in[8];  const float* be0 = (const float*)d_in[9];
    const float* wg1 = (const float*)d_in[10]; const float* bg1 = (const float*)d_in[11];
    const float* W1  = (const float*)d_in[12]; const float* as1w= (const float*)d_in[13];
    const float* ad1w= (const float*)d_in[14]; const float* bb1 = (const float*)d_in[15];
    const float* g1  = (const float*)d_in[16]; const float* be1 = (const float*)d_in[17];
    const float* wg2 = (const float*)d_in[18]; const float* bg2 = (const float*)d_in[19];
    const float* W2  = (const float*)d_in[20]; const float* as2w= (const float*)d_in[21];
    const float* ad2w= (const float*)d_in[22]; const float* bb2 = (const float*)d_in[23];
    const float* p1w = (const float*)d_in[24]; const float* p1b = (const float*)d_in[25];
    const float* p2w = (const float*)d_in[26]; const float* p2b = (const float*)d_in[27];
    const float* p3w = (const float*)d_in[28]; const float* p3b = (const float*)d_in[29];
    const float* p4w = (const float*)d_in[30]; const float* p4b = (const float*)d_in[31];

    const int N  = in_sizes[0] / 32;
    const int E  = in_sizes[1] / 2;
    const int Ep = E + N;                       // edges + self-loops

    float* dout  = (float*)d_out;               // xf: [N,320] row-major
    float* z_out = dout + (size_t)N * 320;      // z:  [N,1]

    // workspace layout (floats)
    float* ws = (float*)d_ws;
    size_t o = 0;
    auto alloc = [&](size_t n) { float* p = ws + o; o += n; return p; };
    float* dinv = alloc((size_t)N);
    float* h    = alloc((size_t)N * 32);
    float* bufA = alloc((size_t)N * 64);        // t1 / xw1 / z3
    float* bufB = alloc((size_t)N * 64);        // t2 / gcn1out
    float* as_  = alloc((size_t)N * HEADS);
    float* ad_  = alloc((size_t)N * HEADS);
    float* mseg = alloc((size_t)N * HEADS);
    float* sseg = alloc((size_t)N * HEADS);
    float* eb   = alloc((size_t)Ep * HEADS);
    float* xl1  = alloc((size_t)N * 320);       // gat1 features / z1
    float* acc1 = alloc((size_t)N * 64);
    float* xw2  = alloc((size_t)N * 128);       // gcn2 xw / z2
    float* g2o  = alloc((size_t)N * 128);       // gcn2 aggregated
    float* xl2  = alloc((size_t)N * 1120);      // gat2 features
    float* acc2 = alloc((size_t)N * 224);
    (void)ws_size; (void)n_in; (void)out_size;

    // degree^-1/2 (self-loops counted via init=1)
    fill_f32<<<B256(N), 256, 0, stream>>>(dinv, 1.0f, (size_t)N);
    deg_accum<<<B256(E), 256, 0, stream>>>(ei, E, dinv);
    deg_to_dinv<<<B256(N), 256, 0, stream>>>(dinv, N);

    // nn1 MLP
    gemm(1, x, 32, w1, 64, b1, bufA, 64, N, 32, 64, stream);
    gemm(1, bufA, 64, w2, 64, b2, bufB, 64, N, 64, 64, stream);
    gemm(0, bufB, 64, w3, 32, b3, h, 32, N, 64, 32, stream);
    // bn0 + lrelu -> h (in place) and xf cols [0,32)
    bn_lrelu<<<32, 256, 0, stream>>>(h, 32, g0, be0, h, 32, dout, 320, N);

    // gcn1: xw = h@Wg1; out = scatter + bias
    gemm(0, h, 32, wg1, 64, nullptr, bufA, 64, N, 32, 64, stream);
    init_bias2d<<<B256((size_t)N * 64), 256, 0, stream>>>(bufB, bg1, 63, (size_t)N * 64);
    gcn_scatter<<<B256((size_t)Ep * 64), 256, 0, stream>>>(ei, E, Ep, dinv, bufA, bufB, 6);

    // gat1 (C=64)
    gemm(0, bufB, 64, W1, 320, nullptr, xl1, 320, N, 64, 320, stream);
    gat_attn<<<B256(N * HEADS), 256, 0, stream>>>(xl1, as1w, ad1w, as_, ad_, N, 64);
    fill_f32<<<B256((size_t)N * HEADS), 256, 0, stream>>>(mseg, -INFINITY, (size_t)N * HEADS);
    fill_f32<<<B256((size_t)N * HEADS), 256, 0, stream>>>(sseg, 0.f, (size_t)N * HEADS);
    fill_f32<<<B256((size_t)N * 64), 256, 0, stream>>>(acc1, 0.f, (size_t)N * 64);
    gat_e1<<<B256(Ep * HEADS), 256, 0, stream>>>(ei, E, Ep, as_, ad_, eb, mseg);
    gat_e2<<<B256(Ep * HEADS), 256, 0, stream>>>(ei, E, Ep, eb, mseg, sseg);
    gat_e3<<<B256((size_t)Ep * 64), 256, 0, stream>>>(ei, E, Ep, eb, sseg, xl1, acc1, 64);
    gat_fin<<<B256((size_t)N * 64), 256, 0, stream>>>(acc1, bb1, acc1, 64, 64, N, 0);
    // bn1 + lrelu -> xf cols [32,96)
    bn_lrelu<<<64, 256, 0, stream>>>(acc1, 64, g1, be1, dout + 32, 320, nullptr, 0, N);

    // gcn2 on skip1 = xf[:, 0:96]
    gemm(0, dout, 320, wg2, 128, nullptr, xw2, 128, N, 96, 128, stream);
    init_bias2d<<<B256((size_t)N * 128), 256, 0, stream>>>(g2o, bg2, 127, (size_t)N * 128);
    gcn_scatter<<<B256((size_t)Ep * 128), 256, 0, stream>>>(ei, E, Ep, dinv, xw2, g2o, 7);

    // gat2 (C=224) -> lrelu -> xf cols [96,320)
    gemm(0, g2o, 128, W2, 1120, nullptr, xl2, 1120, N, 128, 1120, stream);
    gat_attn<<<B256(N * HEADS), 256, 0, stream>>>(xl2, as2w, ad2w, as_, ad_, N, 224);
    fill_f32<<<B256((size_t)N * HEADS), 256, 0, stream>>>(mseg, -INFINITY, (size_t)N * HEADS);
    fill_f32<<<B256((size_t)N * HEADS), 256, 0, stream>>>(sseg, 0.f, (size_t)N * HEADS);
    fill_f32<<<B256((size_t)N * 224), 256, 0, stream>>>(acc2, 0.f, (size_t)N * 224);
    gat_e1<<<B256(Ep * HEADS), 256, 0, stream>>>(ei, E, Ep, as_, ad_, eb, mseg);
    gat_e2<<<B256(Ep * HEADS), 256, 0, stream>>>(ei, E, Ep, eb, mseg, sseg);
    gat_e3<<<B256((size_t)Ep * 224), 256, 0, stream>>>(ei, E, Ep, eb, sseg, xl2, acc2, 224);
    gat_fin<<<B256((size_t)N * 224), 256, 0, stream>>>(acc2, bb2, dout + 96, 320, 224, N, 1);

    // node_pred MLP on xf
    gemm(1, dout, 320, p1w, 160, p1b, xl1, 160, N, 320, 160, stream);
    gemm(1, xl1, 160, p2w, 80, p2b, xw2, 80, N, 160, 80, stream);
    gemm(1, xw2, 80, p3w, 40, p3b, bufA, 40, N, 80, 40, stream);
    gemm(2, bufA, 40, p4w, 1, p4b, z_out, 1, N, 40, 1, stream);
}